// GCNN_23510650978666
// MI455X (gfx1250) — compile-verified
//
#include <hip/hip_runtime.h>
#include <math.h>

#define KK 1024
#define HH 64
#define CC 4
#define BN_EPS 1e-3f
#define KC 128           // k-slice staged in LDS per iteration
#define YT_STRIDE 130    // padded LDS row stride (floats): keeps k-pairs 8B aligned

typedef float v2f __attribute__((ext_vector_type(2)));
typedef float v8f __attribute__((ext_vector_type(8)));

// ---------------------------------------------------------------------------
// K1: adjacency rows  A[b,i,:] = softmax_j( exp(-dist(i,j) * prec) )
// one 256-thread block per (i, b); each thread covers 4 columns
// ---------------------------------------------------------------------------
__global__ void __launch_bounds__(256)
adj_kernel(const float* __restrict__ coords, const float* __restrict__ prec_ptr,
           float* __restrict__ A) {
  const int i = blockIdx.x;
  const int b = blockIdx.y;
  const int tid = threadIdx.x;
  const float p = prec_ptr[0];
  const float* cb = coords + (size_t)b * KK * 3;
  const float cix = cb[i * 3 + 0], ciy = cb[i * 3 + 1], ciz = cb[i * 3 + 2];
  const float ni = cix * cix + ciy * ciy + ciz * ciz;

  float e[4];
  float m = -1e30f;
#pragma unroll
  for (int t = 0; t < 4; ++t) {
    const int j = tid + t * 256;
    const float cjx = cb[j * 3 + 0], cjy = cb[j * 3 + 1], cjz = cb[j * 3 + 2];
    const float nj = cjx * cjx + cjy * cjy + cjz * cjz;
    const float dist = ni - 2.0f * (cix * cjx + ciy * cjy + ciz * cjz) + nj;
    const float a = expf(-dist * p);
    e[t] = a;
    m = fmaxf(m, a);
  }

  __shared__ float red[256];
  red[tid] = m;
  __syncthreads();
  for (int s = 128; s > 0; s >>= 1) {
    if (tid < s) red[tid] = fmaxf(red[tid], red[tid + s]);
    __syncthreads();
  }
  const float M = red[0];
  __syncthreads();

  float ssum = 0.0f;
#pragma unroll
  for (int t = 0; t < 4; ++t) {
    e[t] = expf(e[t] - M);
    ssum += e[t];
  }
  red[tid] = ssum;
  __syncthreads();
  for (int s = 128; s > 0; s >>= 1) {
    if (tid < s) red[tid] += red[tid + s];
    __syncthreads();
  }
  const float inv = 1.0f / red[0];

  float* Ar = A + ((size_t)b * KK + i) * KK;
#pragma unroll
  for (int t = 0; t < 4; ++t) Ar[tid + t * 256] = e[t] * inv;
}

// ---------------------------------------------------------------------------
// K1b: column means of A  (collapsed final block needs only these)
// ---------------------------------------------------------------------------
__global__ void __launch_bounds__(256)
colmean_kernel(const float* __restrict__ A, float* __restrict__ cm) {
  const int j = blockIdx.x * 256 + threadIdx.x;
  const int b = blockIdx.y;
  const float* Ab = A + (size_t)b * KK * KK;
  float s = 0.0f;
  for (int mrow = 0; mrow < KK; ++mrow) s += Ab[(size_t)mrow * KK + j];
  cm[b * KK + j] = s * (1.0f / KK);
}

// ---------------------------------------------------------------------------
// K2: y = x @ W_top, z = x @ W_bot   (small dense projections, per row)
// ---------------------------------------------------------------------------
template <int FIN>
__global__ void __launch_bounds__(256)
yz_kernel(const float* __restrict__ x, const float* __restrict__ W,
          float* __restrict__ y, float* __restrict__ z) {
  const int idx = blockIdx.x * 256 + threadIdx.x;
  const int n = idx & (HH - 1);
  const int row = idx >> 6;  // b*K + k
  const float* xr = x + (size_t)row * FIN;
  float yv = 0.0f, zv = 0.0f;
#pragma unroll 8
  for (int f = 0; f < FIN; ++f) {
    const float xv = xr[f];
    yv += xv * W[f * HH + n];
    zv += xv * W[(FIN + f) * HH + n];
  }
  y[(size_t)row * HH + n] = yv;
  z[(size_t)row * HH + n] = zv;
}

// ---------------------------------------------------------------------------
// K3: h = A @ y  (1024x1024 x 1024x64, fp32 WMMA) with LDS-staged Y.
//   * Y slice [kc:kc+128, 0:64] is cooperatively transposed into LDS
//     (Yt[n][k], stride 130) so each B fragment is one ds_load_b64.
//   * A fragments stream from global (read exactly once, L2 resident).
//   * Fused epilogue: bn = (h + z + bias)*(gamma/sqrt(1+eps)) + beta;
//     out[:,0:64] = bn ; out[:,64:128] = relu(bn)
// 256 threads = 8 waves; wave owns a 16x64 strip; grid.x = B * 8
// ---------------------------------------------------------------------------
__global__ void __launch_bounds__(256)
agemm_kernel(const float* __restrict__ A, const float* __restrict__ Y,
             const float* __restrict__ Z, const float* __restrict__ bias,
             const float* __restrict__ gamma, const float* __restrict__ beta,
             float* __restrict__ out) {
  __shared__ float Yt[HH * YT_STRIDE];  // 64 x 130 floats = 33.3 KB

  const int wave = threadIdx.x >> 5;
  const int lane = threadIdx.x & 31;
  const int b = blockIdx.x >> 3;
  const int blk = blockIdx.x & 7;
  const int m0 = blk * 128 + wave * 16;

  const int koff = (lane < 16) ? 0 : 2;  // fp32 A-frag: lane halves hold K 0/1 vs 2/3
  const int nlo = lane & 15;

  const float* Ab = A + ((size_t)b * KK + m0 + nlo) * KK;  // this lane's A row
  const float* Yb = Y + (size_t)b * KK * HH;

  // transpose-loader indices (all 256 threads)
  const int ln0 = (threadIdx.x & 15) * 4;  // 4 consecutive n per thread
  const int lk0 = threadIdx.x >> 4;        // 16 k-rows per pass

  v8f acc0 = 0.0f, acc1 = 0.0f, acc2 = 0.0f, acc3 = 0.0f;

  for (int kc = 0; kc < KK; kc += KC) {
    // ---- cooperative transposed stage of Y[kc:kc+KC, :] into LDS ----
    const float* src = Yb + (size_t)kc * HH;
#pragma unroll
    for (int kk = 0; kk < KC; kk += 16) {
      const float4 v = *(const float4*)(src + (size_t)(kk + lk0) * HH + ln0);
      const int kl = kk + lk0;
      Yt[(ln0 + 0) * YT_STRIDE + kl] = v.x;
      Yt[(ln0 + 1) * YT_STRIDE + kl] = v.y;
      Yt[(ln0 + 2) * YT_STRIDE + kl] = v.z;
      Yt[(ln0 + 3) * YT_STRIDE + kl] = v.w;
    }
    __syncthreads();

    // ---- WMMA over the staged slice ----
#pragma unroll 4
    for (int k = 0; k < KC; k += 4) {
      const v2f a = *(const v2f*)(Ab + kc + k + koff);
      const int kl = k + koff;
      const v2f b0 = *(const v2f*)(&Yt[(nlo + 0) * YT_STRIDE + kl]);
      const v2f b1 = *(const v2f*)(&Yt[(nlo + 16) * YT_STRIDE + kl]);
      const v2f b2 = *(const v2f*)(&Yt[(nlo + 32) * YT_STRIDE + kl]);
      const v2f b3 = *(const v2f*)(&Yt[(nlo + 48) * YT_STRIDE + kl]);
      acc0 = __builtin_amdgcn_wmma_f32_16x16x4_f32(false, a, false, b0, (short)0, acc0, false, false);
      acc1 = __builtin_amdgcn_wmma_f32_16x16x4_f32(false, a, false, b1, (short)0, acc1, false, false);
      acc2 = __builtin_amdgcn_wmma_f32_16x16x4_f32(false, a, false, b2, (short)0, acc2, false, false);
      acc3 = __builtin_amdgcn_wmma_f32_16x16x4_f32(false, a, false, b3, (short)0, acc3, false, false);
    }
    __syncthreads();
  }

  const float bb = bias[0];
  const float invs = rsqrtf(1.0f + BN_EPS);
  const int mbase = m0 + ((lane >> 4) << 3);  // C/D layout: lanes 16-31 hold M=r+8

  v8f accs[4] = {acc0, acc1, acc2, acc3};
#pragma unroll
  for (int t = 0; t < 4; ++t) {
    const int col = t * 16 + nlo;
    const float sc = gamma[col] * invs;
    const float bt = beta[col];
#pragma unroll
    for (int r = 0; r < 8; ++r) {
      const int mrow = mbase + r;
      const float h = accs[t][r] + Z[((size_t)b * KK + mrow) * HH + col] + bb;
      const float bn = h * sc + bt;
      float* orow = out + ((size_t)b * KK + mrow) * 128;
      orow[col] = bn;
      orow[64 + col] = fmaxf(bn, 0.0f);
    }
  }
}

// ---------------------------------------------------------------------------
// K4: collapsed final block reductions:
//   u[b,f]  = sum_m cm[b,m] * x3[b,m,f] ;  xm[b,f] = mean_m x3[b,m,f]
// ---------------------------------------------------------------------------
__global__ void __launch_bounds__(128)
final_reduce_kernel(const float* __restrict__ x3, const float* __restrict__ cm,
                    float* __restrict__ u, float* __restrict__ xm) {
  const int b = blockIdx.x;
  const int f = threadIdx.x;
  const float* xb = x3 + (size_t)b * KK * 128;
  const float* cb = cm + b * KK;
  float us = 0.0f, ms = 0.0f;
  for (int mrow = 0; mrow < KK; ++mrow) {
    const float v = xb[(size_t)mrow * 128 + f];
    us += cb[mrow] * v;
    ms += v;
  }
  u[b * 128 + f] = us;
  xm[b * 128 + f] = ms * (1.0f / KK);
}

// ---------------------------------------------------------------------------
// K5: logits + softmax head:  out[b,c] = softmax_c(u@W3_top + xm@W3_bot + b3)
// ---------------------------------------------------------------------------
__global__ void __launch_bounds__(128)
head_kernel(const float* __restrict__ u, const float* __restrict__ xm,
            const float* __restrict__ W3, const float* __restrict__ b3,
            float* __restrict__ out) {
  const int tid = threadIdx.x;
  const int b = tid >> 2;
  const int c = tid & 3;
  float s = b3[0];
  for (int f = 0; f < 128; ++f) {
    s += u[b * 128 + f] * W3[f * CC + c] + xm[b * 128 + f] * W3[(128 + f) * CC + c];
  }
  __shared__ float logit[128];
  logit[tid] = s;
  __syncthreads();
  const int base = b << 2;
  const float m = fmaxf(fmaxf(logit[base], logit[base + 1]),
                        fmaxf(logit[base + 2], logit[base + 3]));
  float sum = 0.0f;
#pragma unroll
  for (int cc = 0; cc < 4; ++cc) sum += expf(logit[base + cc] - m);
  out[b * CC + c] = expf(s - m) / sum;
}

// ---------------------------------------------------------------------------
extern "C" void kernel_launch(void* const* d_in, const int* in_sizes, int n_in,
                              void* d_out, int out_size, void* d_ws, size_t ws_size,
                              hipStream_t stream) {
  const int B = 32;
  const float* x      = (const float*)d_in[0];
  const float* coords = (const float*)d_in[1];
  const float* prec   = (const float*)d_in[2];
  const float* W0 = (const float*)d_in[3];
  const float* W1 = (const float*)d_in[4];
  const float* W2 = (const float*)d_in[5];
  const float* W3 = (const float*)d_in[6];
  const float* b0 = (const float*)d_in[7];
  const float* b1 = (const float*)d_in[8];
  const float* b2 = (const float*)d_in[9];
  const float* b3 = (const float*)d_in[10];
  const float* g0  = (const float*)d_in[11];
  const float* be0 = (const float*)d_in[12];
  const float* g1  = (const float*)d_in[13];
  const float* be1 = (const float*)d_in[14];
  const float* g2  = (const float*)d_in[15];
  const float* be2 = (const float*)d_in[16];
  float* out = (float*)d_out;

  // workspace carve-up (all fp32 elements), ~185 MB total
  char* ws = (char*)d_ws;
  size_t off = 0;
  auto alloc = [&](size_t bytes) {
    void* p = ws + off;
    off = (off + bytes + 255) & ~(size_t)255;
    return p;
  };
  float* A     = (float*)alloc((size_t)B * KK * KK * 4);   // 134.2 MB
  float* cm    = (float*)alloc((size_t)B * KK * 4);
  float* y     = (float*)alloc((size_t)B * KK * HH * 4);   // 8.4 MB
  float* z     = (float*)alloc((size_t)B * KK * HH * 4);   // 8.4 MB
  float* xbuf0 = (float*)alloc((size_t)B * KK * 128 * 4);  // 16.8 MB
  float* xbuf1 = (float*)alloc((size_t)B * KK * 128 * 4);  // 16.8 MB
  float* u     = (float*)alloc((size_t)B * 128 * 4);
  float* xmean = (float*)alloc((size_t)B * 128 * 4);
  (void)ws_size; (void)in_sizes; (void)n_in; (void)out_size;

  // 1) adjacency (softmax rows) + column means
  adj_kernel<<<dim3(KK, B), 256, 0, stream>>>(coords, prec, A);
  colmean_kernel<<<dim3(4, B), 256, 0, stream>>>(A, cm);

  const int yzBlocks = (B * KK * HH) / 256;  // 8192
  const int gemmBlocks = B * 8;              // 256

  // 2) block 0: x[.,16] -> xbuf0[.,128]
  yz_kernel<16><<<yzBlocks, 256, 0, stream>>>(x, W0, y, z);
  agemm_kernel<<<gemmBlocks, 256, 0, stream>>>(A, y, z, b0, g0, be0, xbuf0);

  // 3) block 1: xbuf0 -> xbuf1
  yz_kernel<128><<<yzBlocks, 256, 0, stream>>>(xbuf0, W1, y, z);
  agemm_kernel<<<gemmBlocks, 256, 0, stream>>>(A, y, z, b1, g1, be1, xbuf1);

  // 4) block 2: xbuf1 -> xbuf0
  yz_kernel<128><<<yzBlocks, 256, 0, stream>>>(xbuf1, W2, y, z);
  agemm_kernel<<<gemmBlocks, 256, 0, stream>>>(A, y, z, b2, g2, be2, xbuf0);

  // 5) final block collapsed through the mean over K
  final_reduce_kernel<<<B, 128, 0, stream>>>(xbuf0, cm, u, xmean);
  head_kernel<<<1, 128, 0, stream>>>(u, xmean, W3, b3, out);
}